// Beta_LR_41308995453190
// MI455X (gfx1250) — compile-verified
//
#include <hip/hip_runtime.h>

#define B_    8
#define O_    4
#define L_    1024
#define E_    1024
#define BETA_ 512
#define NSEG_ 12
#define NG_   (B_ * O_)          // 32 (b,o) groups
#define NROW_ 13                 // feature rows per group: 10 ctx + q + o + allc

typedef __attribute__((ext_vector_type(2))) float v2f;
typedef __attribute__((ext_vector_type(8))) float v8f;

// ---------------------------------------------------------------------------
// Kernel 1: fused "cumsum at cuts" -> segment means -> feature matrix X
// X rows per group g=(b*O+o): r=0..9 ctx_r, r=10 q, r=11 o, r=12 allc.
// grid = NG_*4 blocks x 256 threads; one thread per embedding column e.
// Inner loop bounds are block-uniform (idx depends only on b) -> no divergence,
// tiny register footprint -> high occupancy for MLP over the serial add chain.
// Explicit prefetch (global_prefetch_b8) runs 16 L-rows ahead of the stream.
// ---------------------------------------------------------------------------
__global__ void seg_feats_kernel(const float* __restrict__ hidden,
                                 const int* __restrict__ idx,
                                 float* __restrict__ X) {
  int bo = blockIdx.x >> 2;                              // 0..31
  int b  = bo / O_;
  int e  = ((blockIdx.x & 3) << 8) + threadIdx.x;        // 0..1023

  int pos[13];
  pos[0] = 1;
#pragma unroll
  for (int k = 0; k < NSEG_; ++k) pos[k + 1] = idx[b * NSEG_ + k];

  const float* base = hidden + (size_t)bo * L_ * E_ + e;

  float seg[12];                                         // seg[k] = P[pos[k+1]] - P[pos[k]]
  int l = pos[0];
#pragma unroll 1
  for (int k = 0; k < 12; ++k) {
    float s = 0.f;
    int end = pos[k + 1];
#pragma unroll 1
    for (; l < end; ++l) {
      __builtin_prefetch(base + (size_t)(l + 16) * E_, 0, 3);
      s += base[(size_t)l * E_];
    }
    seg[k] = s;
  }

  float* xg = X + (size_t)bo * NROW_ * E_ + e;
#pragma unroll
  for (int r = 0; r < 10; ++r)
    xg[r * E_] = seg[r] / (float)(pos[r + 1] - pos[r]);
  xg[10 * E_] = seg[10] / (float)(pos[11] - pos[10]);    // q
  xg[11 * E_] = seg[11] / (float)(pos[12] - pos[11]);    // o
  float allsum = 0.f;
#pragma unroll
  for (int k = 0; k < 10; ++k) allsum += seg[k];
  xg[12 * E_] = allsum / (float)(pos[10] - pos[0]);      // allc
}

// ---------------------------------------------------------------------------
// Generic fp32 WMMA GEMM: C = act(A[MxK] @ W[KxN] + bias[N])
// ACT: 0 = identity, 1 = relu, 2 = 1 + relu  (the _get_param epilogue)
//
// 2x2 register blocking: one wave owns a 32x32 output block = four 16x16
// V_WMMA_F32_16X16X4_F32 accumulators. Per K-step (K+=4) it loads 2 A frags
// + 2 B frags (1 KB) and issues 4 WMMAs (8 KFLOP) -> 8 FLOP/B from L2 and
// 4 independent D-chains that pipeline through the XDL without RAW stalls.
// A-frag layout per ISA: lane<16 holds row M=lane, K=k..k+1; lane>=16 row
// M=lane-16, K=k+2..k+3; B symmetric; C/D: VGPR r -> M=r+8*(lane>=16),
// N=lane&15. Early-exit is wave-uniform so EXEC is all-ones at every WMMA.
// All call sites have M%32==0 and N%32==0.
// ---------------------------------------------------------------------------
template <int ACT>
__global__ void gemm_wmma_f32(const float* __restrict__ A,
                              const float* __restrict__ W,
                              const float* __restrict__ bias,
                              float* __restrict__ C,
                              int M, int N, int K) {
  int wave = (int)((blockIdx.x * blockDim.x + threadIdx.x) >> 5);
  int lane = threadIdx.x & 31;
  int ntiles = N >> 5;                       // 32-wide column blocks
  int tiles  = (M >> 5) * ntiles;
  if (wave >= tiles) return;                 // uniform per wave

  int tm  = wave / ntiles;
  int tn  = wave - tm * ntiles;
  int r16 = lane & 15;
  int kh  = (lane >> 4) << 1;                // 0 or 2

  const float* arow0 = A + (size_t)(tm * 32 + r16) * K + kh;        // rows +0..15
  const float* arow1 = arow0 + (size_t)16 * K;                      // rows +16..31
  const float* wcol0 = W + (size_t)kh * N + tn * 32 + r16;          // cols +0..15
  const float* wcol1 = wcol0 + 16;                                  // cols +16..31

  v8f acc00 = {}, acc01 = {}, acc10 = {}, acc11 = {};
  for (int k = 0; k < K; k += 4) {
    v2f a0, a1, b0, b1;
    a0.x = arow0[k];
    a0.y = arow0[k + 1];
    a1.x = arow1[k];
    a1.y = arow1[k + 1];
    b0.x = wcol0[(size_t)k * N];
    b0.y = wcol0[(size_t)(k + 1) * N];
    b1.x = wcol1[(size_t)k * N];
    b1.y = wcol1[(size_t)(k + 1) * N];
    acc00 = __builtin_amdgcn_wmma_f32_16x16x4_f32(false, a0, false, b0,
                                                  (short)0, acc00, false, false);
    acc01 = __builtin_amdgcn_wmma_f32_16x16x4_f32(false, a0, false, b1,
                                                  (short)0, acc01, false, false);
    acc10 = __builtin_amdgcn_wmma_f32_16x16x4_f32(false, a1, false, b0,
                                                  (short)0, acc10, false, false);
    acc11 = __builtin_amdgcn_wmma_f32_16x16x4_f32(false, a1, false, b1,
                                                  (short)0, acc11, false, false);
  }

  int orow = tm * 32 + ((lane >> 4) << 3);   // +0 or +8 within the 16-row tile
  int ocol = tn * 32 + r16;
  float bb0 = bias[ocol];
  float bb1 = bias[ocol + 16];
#pragma unroll
  for (int r = 0; r < 8; ++r) {
    float v00 = acc00[r] + bb0;
    float v01 = acc01[r] + bb1;
    float v10 = acc10[r] + bb0;
    float v11 = acc11[r] + bb1;
    if (ACT >= 1) {
      v00 = fmaxf(v00, 0.f); v01 = fmaxf(v01, 0.f);
      v10 = fmaxf(v10, 0.f); v11 = fmaxf(v11, 0.f);
    }
    if (ACT == 2) { v00 += 1.f; v01 += 1.f; v10 += 1.f; v11 += 1.f; }
    C[(size_t)(orow + r) * N + ocol]           = v00;
    C[(size_t)(orow + r) * N + ocol + 16]      = v01;
    C[(size_t)(orow + 16 + r) * N + ocol]      = v10;
    C[(size_t)(orow + 16 + r) * N + ocol + 16] = v11;
  }
}

// ---------------------------------------------------------------------------
// Pool 1: softmax over the 10 ctx rows (logits s1 rows are (g*13+r), N=512),
// then ia/ib weighted sums of param halves -> iab (32 x 1024).
// ---------------------------------------------------------------------------
__global__ void pool1_softmax(const float* __restrict__ s1,
                              const float* __restrict__ param,
                              float* __restrict__ iab) {
  int t = blockIdx.x * blockDim.x + threadIdx.x;
  if (t >= NG_ * BETA_) return;
  int g = t >> 9, n = t & 511;
  const float* sg = s1 + (size_t)g * NROW_ * BETA_ + n;
  float lg[10], m = -3.4e38f;
#pragma unroll
  for (int r = 0; r < 10; ++r) { lg[r] = sg[r * BETA_]; m = fmaxf(m, lg[r]); }
  float sum = 0.f;
#pragma unroll
  for (int r = 0; r < 10; ++r) { lg[r] = __expf(lg[r] - m); sum += lg[r]; }
  float inv = 1.f / sum;
  const float* pg = param + (size_t)g * NROW_ * E_ + n;
  float ia = 0.f, ib = 0.f;
#pragma unroll
  for (int r = 0; r < 10; ++r) {
    float w = lg[r] * inv;
    ia += w * pg[r * E_];
    ib += w * pg[r * E_ + BETA_];
  }
  iab[g * E_ + n]         = ia;
  iab[g * E_ + BETA_ + n] = ib;
}

// Build X2 (640 x 1024): rows ((g*10+j)*2+t): t=0 -> param row (g*13+j), t=1 -> iab[g]
__global__ void build_x2(const float* __restrict__ param,
                         const float* __restrict__ iab,
                         float* __restrict__ X2) {
  int t = blockIdx.x * blockDim.x + threadIdx.x;
  if (t >= 640 * E_) return;
  int i   = t & 1023;
  int rid = t >> 10;
  int tt  = rid & 1;
  int gj  = rid >> 1;
  int g = gj / 10, j = gj - g * 10;
  X2[t] = tt ? iab[g * E_ + i] : param[(size_t)(g * NROW_ + j) * E_ + i];
}

// Pool 2: softmax over the stacked pair (2), na/nb; emit X3 = 1/na | 1/nb (320 x 1024)
__global__ void pool2_softmax(const float* __restrict__ s2,
                              const float* __restrict__ param,
                              const float* __restrict__ iab,
                              float* __restrict__ X3) {
  int t = blockIdx.x * blockDim.x + threadIdx.x;
  if (t >= 320 * BETA_) return;
  int n  = t & 511;
  int gj = t >> 9;
  int g = gj / 10, j = gj - g * 10;
  const float* sg = s2 + (size_t)(gj * 2) * BETA_ + n;
  float l0 = sg[0], l1 = sg[BETA_];
  float m = fmaxf(l0, l1);
  float e0 = __expf(l0 - m), e1 = __expf(l1 - m);
  float inv = 1.f / (e0 + e1);
  float w0 = e0 * inv, w1 = e1 * inv;
  float a0 = param[(size_t)(g * NROW_ + j) * E_ + n];
  float b0 = param[(size_t)(g * NROW_ + j) * E_ + BETA_ + n];
  float ia = iab[g * E_ + n], ib = iab[g * E_ + BETA_ + n];
  X3[(size_t)gj * E_ + n]         = 1.f / (w0 * a0 + w1 * ia);
  X3[(size_t)gj * E_ + BETA_ + n] = 1.f / (w0 * b0 + w1 * ib);
}

// Pool 3: softmax over 10 (s3 rows are gj = g*10+j), ua/ub, reciprocal -> cat cols 0..1023
__global__ void pool3_softmax(const float* __restrict__ s3,
                              const float* __restrict__ X3,
                              float* __restrict__ cat) {
  int t = blockIdx.x * blockDim.x + threadIdx.x;
  if (t >= NG_ * BETA_) return;
  int g = t >> 9, n = t & 511;
  const float* sg = s3 + (size_t)g * 10 * BETA_ + n;
  float lg[10], m = -3.4e38f;
#pragma unroll
  for (int r = 0; r < 10; ++r) { lg[r] = sg[r * BETA_]; m = fmaxf(m, lg[r]); }
  float sum = 0.f;
#pragma unroll
  for (int r = 0; r < 10; ++r) { lg[r] = __expf(lg[r] - m); sum += lg[r]; }
  float inv = 1.f / sum;
  float ua = 0.f, ub = 0.f;
#pragma unroll
  for (int r = 0; r < 10; ++r) {
    float w = lg[r] * inv;
    ua += w * X3[(size_t)(g * 10 + r) * E_ + n];
    ub += w * X3[(size_t)(g * 10 + r) * E_ + BETA_ + n];
  }
  cat[(size_t)g * 4096 + n]         = 1.f / ua;
  cat[(size_t)g * 4096 + BETA_ + n] = 1.f / ub;
}

// cat cols 1024..4095 = param rows 12 (a_ac|b_ac), 11 (a_o|b_o), 10 (a_q|b_q)
__global__ void cat_tail(const float* __restrict__ param, float* __restrict__ cat) {
  int t = blockIdx.x * blockDim.x + threadIdx.x;
  if (t >= NG_ * 3072) return;
  int g = t / 3072, i = t - g * 3072;
  int s = i >> 10, i2 = i & 1023;
  cat[(size_t)g * 4096 + 1024 + i] = param[(size_t)(g * NROW_ + (12 - s)) * E_ + i2];
}

// out[g] = hfin[g,:] . Wl + bl  (fixed-order LDS reduction -> deterministic)
__global__ void final_out_kernel(const float* __restrict__ hfin,
                                 const float* __restrict__ Wl,
                                 const float* __restrict__ bl,
                                 float* __restrict__ out) {
  __shared__ float red[256];
  int g = blockIdx.x;
  float p = 0.f;
  for (int n = threadIdx.x; n < BETA_; n += 256)
    p += hfin[(size_t)g * BETA_ + n] * Wl[n];
  red[threadIdx.x] = p;
  __syncthreads();
  for (int s = 128; s > 0; s >>= 1) {
    if ((int)threadIdx.x < s) red[threadIdx.x] += red[threadIdx.x + s];
    __syncthreads();
  }
  if (threadIdx.x == 0) out[g] = red[0] + bl[0];
}

// ---------------------------------------------------------------------------
extern "C" void kernel_launch(void* const* d_in, const int* in_sizes, int n_in,
                              void* d_out, int out_size, void* d_ws, size_t ws_size,
                              hipStream_t stream) {
  const float* hidden = (const float*)d_in[0];
  const int*   idx    = (const int*)  d_in[1];
  const float* Wp  = (const float*)d_in[2];
  const float* bp  = (const float*)d_in[3];
  const float* Wa0 = (const float*)d_in[4];
  const float* ba0 = (const float*)d_in[5];
  const float* Wa  = (const float*)d_in[6];
  const float* ba  = (const float*)d_in[7];
  const float* Wl0 = (const float*)d_in[8];
  const float* bl0 = (const float*)d_in[9];
  const float* Wl  = (const float*)d_in[10];
  const float* bl  = (const float*)d_in[11];
  float* out = (float*)d_out;

  float* ws = (float*)d_ws;
  size_t o = 0;
  float* X     = ws + o; o += (size_t)416 * 1024;   // 13 feature rows per group
  float* param = ws + o; o += (size_t)416 * 1024;   // 1+relu(X@Wp+bp) = [a|b]
  float* h1    = ws + o; o += (size_t)416 * 512;
  float* s1    = ws + o; o += (size_t)416 * 512;
  float* iab   = ws + o; o += (size_t)32  * 1024;   // [ia|ib]
  float* X2    = ws + o; o += (size_t)640 * 1024;
  float* h2    = ws + o; o += (size_t)640 * 512;
  float* s2    = ws + o; o += (size_t)640 * 512;
  float* X3    = ws + o; o += (size_t)320 * 1024;   // [1/na|1/nb]
  float* h3    = ws + o; o += (size_t)320 * 512;
  float* s3    = ws + o; o += (size_t)320 * 512;
  float* cat   = ws + o; o += (size_t)32  * 4096;
  float* hfin  = ws + o; o += (size_t)32  * 512;

  // Stage 1: segment features (the only HBM-heavy pass: <=128MB read once)
  seg_feats_kernel<<<NG_ * 4, 256, 0, stream>>>(hidden, idx, X);

#define GEMM(ACT, Apt, Wpt, Bpt, Cpt, M, N, K)                                  \
  gemm_wmma_f32<ACT><<<((((M) / 32) * ((N) / 32)) + 7) / 8, 256, 0, stream>>>(  \
      Apt, Wpt, Bpt, Cpt, M, N, K)

  // param = 1 + relu(X @ Wp + bp)             (416 x 1024, K=1024)
  GEMM(2, X, Wp, bp, param, 416, 1024, 1024);

  // pool 1 (over 10 ctx segments); concat([a,b]) == full param row
  GEMM(1, param, Wa0, ba0, h1, 416, 512, 1024);
  GEMM(0, h1, Wa, ba, s1, 416, 512, 512);
  pool1_softmax<<<(NG_ * BETA_ + 255) / 256, 256, 0, stream>>>(s1, param, iab);

  // pool 2 (over stacked pair)
  build_x2<<<(640 * E_ + 255) / 256, 256, 0, stream>>>(param, iab, X2);
  GEMM(1, X2, Wa0, ba0, h2, 640, 512, 1024);
  GEMM(0, h2, Wa, ba, s2, 640, 512, 512);
  pool2_softmax<<<(320 * BETA_ + 255) / 256, 256, 0, stream>>>(s2, param, iab, X3);

  // pool 3 (over 10, on reciprocals)
  GEMM(1, X3, Wa0, ba0, h3, 320, 512, 1024);
  GEMM(0, h3, Wa, ba, s3, 320, 512, 512);
  pool3_softmax<<<(NG_ * BETA_ + 255) / 256, 256, 0, stream>>>(s3, X3, cat);

  // head
  cat_tail<<<(NG_ * 3072 + 255) / 256, 256, 0, stream>>>(param, cat);
  GEMM(1, cat, Wl0, bl0, hfin, 32, 512, 4096);
  final_out_kernel<<<NG_, 256, 0, stream>>>(hfin, Wl, bl, out);

#undef GEMM
}